// BiLSTM_CRF_9019431322039
// MI455X (gfx1250) — compile-verified
//
#include <hip/hip_runtime.h>
#include <hip/hip_bf16.h>

// ---------------------------------------------------------------------------
// CRF forward (log-partition) on gfx1250 via scaled exp-space WMMA GEMM.
//
// State per sequence: exp-space probabilities p[48] + scalar logZ, with
// renormalization every RESCALE=4 steps (fp32 dynamic range allows growth
// of ~3e4/step; 4 steps ~1e18 << 3.4e38, so no overflow is possible).
//   v      = E @ p              (E = exp(transitions), WMMA fp32 GEMM)
//   w[i]   = v[i] * exp(feat[i])          -> becomes next p
//   every 4th step: c = max_i w[i]; p <- w/c; logZ += log(c)
// Invariant: exp(alpha[i]) == p[i] * exp(logZ)  (alpha = reference log-state)
// Final: forward = logZ + log( sum_i exp(trans[STOP,i]) * p[i] )
//
// K-permutation trick: the GEMM's K axis is relabeled by a bijection sigma so
// that the B-fragment register pairs coincide exactly with the per-lane state
// registers produced by the previous step's D output — zero cross-lane data
// movement per step (one shfl_xor per rescale is all that remains). The
// permutation is folded into the one-time load of E's columns.
// ---------------------------------------------------------------------------

typedef float v2f __attribute__((ext_vector_type(2)));
typedef float v8f __attribute__((ext_vector_type(8)));

#define N_TAG     48
#define SEQ_LEN   512
#define BATCH     1024
#define START_TAG 46
#define STOP_TAG  47
#define RESCALE   4          // renormalize period (SEQ_LEN % RESCALE == 0)

// raw hardware transcendentals (v_exp_f32 / v_log_f32 / v_rcp_f32); operands
// here never enter the denormal range the libm fixup branches protect.
__device__ __forceinline__ float fast_exp(float x) {
    return __builtin_amdgcn_exp2f(x * 1.44269504088896340736f);
}
__device__ __forceinline__ float fast_log(float x) {
    return __builtin_amdgcn_logf(x) * 0.69314718055994530942f;
}
__device__ __forceinline__ float fast_rcp(float x) {
    return __builtin_amdgcn_rcpf(x);
}

// ---------------------------------------------------------------------------
// Gold path score, one block per sequence, deterministic tree reduction.
// ---------------------------------------------------------------------------
__global__ __launch_bounds__(256)
void crf_gold_kernel(const float* __restrict__ feats,
                     const int*   __restrict__ tags,
                     const float* __restrict__ trans,
                     float*       __restrict__ gold)
{
    const int b = blockIdx.x;
    const int t = threadIdx.x;
    const int*   tg = tags  + (size_t)b * SEQ_LEN;
    const float* fb = feats + (size_t)b * SEQ_LEN * N_TAG;

    float acc = 0.0f;
    for (int s = t; s < SEQ_LEN; s += 256) {
        const int cur  = tg[s];
        const int prev = (s == 0) ? START_TAG : tg[s - 1];
        acc += trans[cur * N_TAG + prev] + fb[(size_t)s * N_TAG + cur];
    }

    __shared__ float red[256];
    red[t] = acc;
    __syncthreads();
    #pragma unroll
    for (int off = 128; off > 0; off >>= 1) {
        if (t < off) red[t] += red[t + off];
        __syncthreads();
    }
    if (t == 0)
        gold[b] = red[0] + trans[STOP_TAG * N_TAG + tg[SEQ_LEN - 1]];
}

// ---------------------------------------------------------------------------
// Forward algorithm, one wave (32 lanes) per 16-sequence tile.
//
// h = lane>>4, sl = lane&15.  Local state index q in 0..23 maps to
//   tag(h, q) = 16*(q/8) + 8*h + (q%8)                (D/C layout of WMMA)
// K-position sigma: K = 4*kb + c  ->  owning half hs = (c >= 2),
//   local index lidx = 2*kb + (c & 1), tag = tag(hs, lidx).
// With this sigma, B-fragment kb == v2f{ p_flat[2kb], p_flat[2kb+1] } on both
// lane halves (B layout assumed to mirror the documented A layout: VGPR0 =
// rows {K, K+2} on lane halves, VGPR1 = rows {K+1, K+3}).
// A-fragment (lane h, sl; tile mt; block kb): E[16*mt+sl][ sigma(4kb+2h..+1) ]
// = E row at the lane's OWN half's tags with local indices {2kb, 2kb+1}.
// ---------------------------------------------------------------------------
__global__ __launch_bounds__(32)
void crf_forward_wmma_kernel(const float* __restrict__ feats,
                             const float* __restrict__ trans,
                             const float* __restrict__ gold,
                             float*       __restrict__ out)
{
    const int lane = threadIdx.x;        // 0..31
    const int h    = lane >> 4;          // lane half (0/1)
    const int sl   = lane & 15;          // seq-in-tile / A-row-in-tile
    const int seq  = blockIdx.x * 16 + sl;

    // ---- loop-invariant A-fragments: E = exp(transitions), sigma-permuted ----
    v2f Ea[3][12];
    #pragma unroll
    for (int mt = 0; mt < 3; ++mt) {
        const int i = 16 * mt + sl;                       // tag_out row
        #pragma unroll
        for (int kb = 0; kb < 12; ++kb) {
            const int l0 = 2 * kb, l1 = 2 * kb + 1;       // local idx of cols
            const int j0 = 16 * (l0 >> 3) + 8 * h + (l0 & 7);  // sigma(4kb+2h)
            const int j1 = 16 * (l1 >> 3) + 8 * h + (l1 & 7);  // sigma(4kb+2h+1)
            Ea[mt][kb].x = fast_exp(trans[i * N_TAG + j0]);
            Ea[mt][kb].y = fast_exp(trans[i * N_TAG + j1]);
        }
    }
    // epilogue weights: exp(trans[STOP, tag]) for this lane's owned tags
    float E2[24];
    #pragma unroll
    for (int q = 0; q < 24; ++q) {
        const int tag = 16 * (q >> 3) + 8 * h + (q & 7);
        E2[q] = fast_exp(trans[STOP_TAG * N_TAG + tag]);
    }

    // ---- state init: p = one-hot(START), logZ = 0 ----
    // pv[kb] holds local indices {2kb, 2kb+1}; it IS B-fragment kb.
    v2f pv[12];
    #pragma unroll
    for (int kb = 0; kb < 12; ++kb) {
        const int t0 = 16 * ((2 * kb) >> 3) + 8 * h + ((2 * kb) & 7);
        const int t1 = 16 * ((2 * kb + 1) >> 3) + 8 * h + ((2 * kb + 1) & 7);
        pv[kb].x = (t0 == START_TAG) ? 1.0f : 0.0f;
        pv[kb].y = (t1 == START_TAG) ? 1.0f : 0.0f;
    }
    float lz = 0.0f;

    const float* fbase = feats + (size_t)seq * SEQ_LEN * N_TAG;

    #pragma unroll 1
    for (int sg = 0; sg < SEQ_LEN; sg += RESCALE) {
        float w[24];

        #pragma unroll
        for (int u = 0; u < RESCALE; ++u) {
            // emissions for this lane's 24 owned tags (3 x 32B, b128 loads);
            // fv[q] matches local index q (tag = 16*(q/8) + 8h + q%8).
            float fv[24];
            const float* fp = fbase + (size_t)(sg + u) * N_TAG + 8 * h;
            #pragma unroll
            for (int mt = 0; mt < 3; ++mt) {
                const float4 lo = *(const float4*)(fp + 16 * mt + 0);
                const float4 hi = *(const float4*)(fp + 16 * mt + 4);
                fv[mt * 8 + 0] = lo.x; fv[mt * 8 + 1] = lo.y;
                fv[mt * 8 + 2] = lo.z; fv[mt * 8 + 3] = lo.w;
                fv[mt * 8 + 4] = hi.x; fv[mt * 8 + 5] = hi.y;
                fv[mt * 8 + 6] = hi.z; fv[mt * 8 + 7] = hi.w;
            }

            // v = E * p via fp32 WMMA; B operands are the state registers.
            #pragma unroll
            for (int mt = 0; mt < 3; ++mt) {
                v8f acc0 = {0.f, 0.f, 0.f, 0.f, 0.f, 0.f, 0.f, 0.f};
                v8f acc1 = {0.f, 0.f, 0.f, 0.f, 0.f, 0.f, 0.f, 0.f};
                #pragma unroll
                for (int kb = 0; kb < 12; kb += 2) {
                    acc0 = __builtin_amdgcn_wmma_f32_16x16x4_f32(
                        false, Ea[mt][kb + 0], false, pv[kb + 0],
                        (short)0, acc0, false, false);
                    acc1 = __builtin_amdgcn_wmma_f32_16x16x4_f32(
                        false, Ea[mt][kb + 1], false, pv[kb + 1],
                        (short)0, acc1, false, false);
                }
                // w = v * exp(feat), still in local-index layout
                #pragma unroll
                for (int r = 0; r < 8; ++r)
                    w[mt * 8 + r] = (acc0[r] + acc1[r]) * fast_exp(fv[mt * 8 + r]);
            }

            if (u < RESCALE - 1) {
                // no rescale: w becomes next state directly
                #pragma unroll
                for (int kb = 0; kb < 12; ++kb) {
                    pv[kb].x = w[2 * kb + 0];
                    pv[kb].y = w[2 * kb + 1];
                }
            }
        }

        // rescale once per group: c = max over all 48 tags of this sequence
        float t12[12];
        #pragma unroll
        for (int q = 0; q < 12; ++q) t12[q] = fmaxf(w[q], w[q + 12]);
        float t6[6];
        #pragma unroll
        for (int q = 0; q < 6; ++q) t6[q] = fmaxf(t12[q], t12[q + 6]);
        float t3[3];
        #pragma unroll
        for (int q = 0; q < 3; ++q) t3[q] = fmaxf(t6[q], t6[q + 3]);
        float c = fmaxf(fmaxf(t3[0], t3[1]), t3[2]);
        c = fmaxf(c, __shfl_xor(c, 16, 32));

        const float rc = fast_rcp(c);
        lz += fast_log(c);
        #pragma unroll
        for (int kb = 0; kb < 12; ++kb) {
            pv[kb].x = w[2 * kb + 0] * rc;
            pv[kb].y = w[2 * kb + 1] * rc;
        }
    }

    // ---- epilogue: forward = logZ + log( sum_i E2[i] * p[i] ) ----
    float part = 0.0f;
    #pragma unroll
    for (int kb = 0; kb < 12; ++kb)
        part += E2[2 * kb] * pv[kb].x + E2[2 * kb + 1] * pv[kb].y;
    part += __shfl_xor(part, 16, 32);

    const float fwd = lz + fast_log(part);
    if (h == 0 && seq < BATCH)
        out[seq] = fwd - gold[seq];
}

// ---------------------------------------------------------------------------
// Launch: gold kernel first, then forward kernel (same stream -> ordered).
// Workspace: gold[BATCH] floats.
// ---------------------------------------------------------------------------
extern "C" void kernel_launch(void* const* d_in, const int* in_sizes, int n_in,
                              void* d_out, int out_size, void* d_ws, size_t ws_size,
                              hipStream_t stream) {
    const float* feats = (const float*)d_in[0];   // [B, S, 48] f32
    const int*   tags  = (const int*)  d_in[1];   // [B, S] int
    const float* trans = (const float*)d_in[2];   // [48, 48] f32
    float*       out   = (float*)d_out;           // [B] f32
    float*       gold  = (float*)d_ws;            // [B] f32 scratch

    crf_gold_kernel<<<BATCH, 256, 0, stream>>>(feats, tags, trans, gold);
    crf_forward_wmma_kernel<<<BATCH / 16, 32, 0, stream>>>(feats, trans, gold, out);
}